// SMAQBlockVQ_17360257810703
// MI455X (gfx1250) — compile-verified
//
#include <hip/hip_runtime.h>

typedef __attribute__((ext_vector_type(2))) float v2f;
typedef __attribute__((ext_vector_type(4))) float v4f;
typedef __attribute__((ext_vector_type(8))) float v8f;

#define NB  16
#define BD  8
#define NC  256
#define DIM 128   // NB * BD

// One workgroup = 16 waves (512 threads). Wave w handles quant block b = w for a
// tile of 16 rows. Centroid scoring runs on the f32 WMMA pipe: per 16-centroid
// tile, 3x V_WMMA_F32_16X16X4_F32 produce acc = dot(shaped, centroid) - c2/2
// directly (the -c2/2 rank-1 term is folded in as a third K=4 WMMA), so the
// VALU only does the running argmax.
__global__ __launch_bounds__(512) void smaq_vq_kernel(
    const float* __restrict__ kin,   // [N, 128]
    const float* __restrict__ Eb,    // [16, 8, 8]
    const float* __restrict__ cent,  // [16, 256, 8]
    const float* __restrict__ dec,   // [16, 256, 8]
    float* __restrict__ out)         // [N, 128]
{
  const int lane = threadIdx.x & 31;
  const int b    = threadIdx.x >> 5;   // wave id == block index
  const int l16  = lane & 15;
  const int half = lane >> 4;          // 0: K=0..1 rows 0..7 ; 1: K=2..3 rows 8..15
  const int e0   = half * 2;           // this lane's K-slot base within a K=4 WMMA
  const int row0 = blockIdx.x * 16;

  // ---- per-lane row slice: x = k[row0 + l16, b*8 .. b*8+8)  (halves duplicate rows)
  const float* krow = kin + (size_t)(row0 + l16) * DIM + b * BD;
  v4f x0 = *(const v4f*)(krow);
  v4f x1 = *(const v4f*)(krow + 4);
  float x[8] = {x0.x, x0.y, x0.z, x0.w, x1.x, x1.y, x1.z, x1.w};

  // ---- shaped[e] = sum_d x[d] * E[b][e][d]  (E wave-uniform -> scalar loads)
  const float* Erow = Eb + b * BD * BD;
  float s[8];
#pragma unroll
  for (int e = 0; e < 8; ++e) {
    float acc = 0.f;
#pragma unroll
    for (int d = 0; d < 8; ++d) acc = fmaf(x[d], Erow[e * BD + d], acc);
    s[e] = acc;
  }

  // ---- A fragments (16x4 f32 = 2 VGPRs; one-time cndmask selects)
  v2f a_lo; a_lo.x = s[e0];     a_lo.y = s[e0 + 1];     // K = 0..3
  v2f a_hi; a_hi.x = s[4 + e0]; a_hi.y = s[4 + e0 + 1]; // K = 4..7
  v2f a_c2; a_c2.x = half ? 0.f : 1.f; a_c2.y = 0.f;    // A3[M,0] = 1

  // running argmax of (dot - c2/2)  ==  argmin of (c2 - 2*dot)  (x2 dropped)
  float best[8];
  int   bestIdx[8];
#pragma unroll
  for (int j = 0; j < 8; ++j) { best[j] = -3.0e38f; bestIdx[j] = 0; }

  // lane's centroid column N = l16; fragment values start at component e0
  const float* cbase = cent + ((size_t)b * NC + l16) * BD + e0;

#pragma unroll 4
  for (int ct = 0; ct < NC / 16; ++ct) {
    // B fragments loaded directly at lane-dependent offset: no selects
    const float* cp = cbase + (size_t)ct * 16 * BD;
    v2f b_lo = *(const v2f*)(cp);      // c[e0], c[e0+1]
    v2f b_hi = *(const v2f*)(cp + 4);  // c[e0+4], c[e0+5]

    // c2: this lane holds 4 of 8 components; xor-16 partner holds the other 4
    float part = b_lo.x * b_lo.x;
    part = fmaf(b_lo.y, b_lo.y, part);
    part = fmaf(b_hi.x, b_hi.x, part);
    part = fmaf(b_hi.y, b_hi.y, part);
    float c2 = part + __shfl_xor(part, 16, 32);

    v2f b_c2; b_c2.x = half ? 0.f : (-0.5f * c2); b_c2.y = 0.f; // B3[0,N] = -c2/2

    // acc[M,N] = -c2[N]/2 + sum_e shaped[M,e]*cent[N,e]   (3x K=4 f32 WMMA)
    v8f acc = {};
    acc = __builtin_amdgcn_wmma_f32_16x16x4_f32(false, a_c2, false, b_c2,
                                                (short)0, acc, false, false);
    acc = __builtin_amdgcn_wmma_f32_16x16x4_f32(false, a_lo, false, b_lo,
                                                (short)0, acc, false, false);
    acc = __builtin_amdgcn_wmma_f32_16x16x4_f32(false, a_hi, false, b_hi,
                                                (short)0, acc, false, false);

    const int myc = ct * 16 + l16;
#pragma unroll
    for (int j = 0; j < 8; ++j) {            // row M = j + 8*half
      bool better = acc[j] > best[j];        // strict '>' keeps lowest ct on ties
      bestIdx[j] = better ? myc : bestIdx[j];
      best[j]    = fmaxf(acc[j], best[j]);
    }
  }

  // ---- argmax-reduce across each 16-lane half (masks < 16 stay in-half)
#pragma unroll
  for (int j = 0; j < 8; ++j) {
#pragma unroll
    for (int mask = 1; mask < 16; mask <<= 1) {
      float os = __shfl_xor(best[j],    mask, 32);
      int   oi = __shfl_xor(bestIdx[j], mask, 32);
      bool better = (os > best[j]) ||
                    (os == best[j] && oi < bestIdx[j]); // first-match like argmin
      best[j]    = better ? os : best[j];
      bestIdx[j] = better ? oi : bestIdx[j];
    }
  }

  // ---- dequantize: row = row0 + j + 8*half; 2 lanes per half store float4s
  const float* dbase = dec + (size_t)b * NC * BD;
#pragma unroll
  for (int j = 0; j < 8; ++j) {
    if (l16 < 2) {
      const int idx = bestIdx[j];                 // uniform within the half
      const int row = row0 + j + 8 * half;
      v4f dv = *(const v4f*)(dbase + (size_t)idx * BD + l16 * 4);
      *(v4f*)(out + (size_t)row * DIM + b * BD + l16 * 4) = dv;
    }
  }
}

extern "C" void kernel_launch(void* const* d_in, const int* in_sizes, int n_in,
                              void* d_out, int out_size, void* d_ws, size_t ws_size,
                              hipStream_t stream) {
  const float* kin  = (const float*)d_in[0];  // (4,16,4096,128)
  const float* Eb   = (const float*)d_in[1];  // (16,8,8)
  const float* cent = (const float*)d_in[2];  // (16,256,8)
  const float* dec  = (const float*)d_in[3];  // (16,256,8)
  float* out = (float*)d_out;

  const int nrows  = in_sizes[0] / DIM;  // 262144
  const int ntiles = nrows / 16;         // 16384 (exact)
  smaq_vq_kernel<<<ntiles, 512, 0, stream>>>(kin, Eb, cent, dec, out);
}